// ElasticLowRankLayer_30279519437306
// MI455X (gfx1250) — compile-verified
//
#include <hip/hip_runtime.h>

// CDNA5 / gfx1250, wave32. D = A(16x4) * B(4x16) + C via V_WMMA_F32_16X16X4_F32.
typedef float v2f __attribute__((ext_vector_type(2)));
typedef float v8f __attribute__((ext_vector_type(8)));

#define N_IN     4096
#define N_OUT    4096
#define RANK     64
#define RES_RANK 4
#define BATCH    8192
#define R_PAD    80   // 68 ranks padded to 5 WMMA M-tiles of 16

// ---------------------------------------------------------------------------
// Prep: build VcatT (R_PAD x N_IN, row-major) = [V | V_res]^T zero-padded,
// and Ucat (N_OUT x R_PAD, row-major) = [U | U_res] zero-padded.
// ---------------------------------------------------------------------------
__global__ void __launch_bounds__(256)
elr_prep(const float* __restrict__ U, const float* __restrict__ V,
         const float* __restrict__ U_res, const float* __restrict__ V_res,
         float* __restrict__ VcatT, float* __restrict__ Ucat) {
  int idx = blockIdx.x * blockDim.x + threadIdx.x;
  const int nV = R_PAD * N_IN;
  if (idx < nV) {
    int r = idx / N_IN;
    int i = idx - r * N_IN;
    float v = 0.0f;
    if (r < RANK)                 v = V[i * RANK + r];
    else if (r < RANK + RES_RANK) v = V_res[i * RES_RANK + (r - RANK)];
    VcatT[idx] = v;
  } else {
    int j = idx - nV;                 // Ucat element
    int o = j / R_PAD;
    int r = j - o * R_PAD;
    float u = 0.0f;
    if (r < RANK)                 u = U[o * RANK + r];
    else if (r < RANK + RES_RANK) u = U_res[o * RES_RANK + (r - RANK)];
    Ucat[j] = u;
  }
}

// ---------------------------------------------------------------------------
// Stage 1: T (R_PAD x BATCH) = VcatT @ x.  One wave = one 16x16 tile of T.
// A-tile (16x4 f32): lane half -> K{+2,+3}, vgpr -> K lsb, M = lane%16.
//   Per lane the two A values are K-contiguous in VcatT row -> one b64 load.
// B-tile (4x16 f32): same K mapping, N = lane%16 -> two 64B-contiguous b32.
// Two accumulators break the dependent-WMMA chain for pipelining.
// ---------------------------------------------------------------------------
__global__ void __launch_bounds__(256)
elr_stage1(const float* __restrict__ x, const float* __restrict__ VcatT,
           float* __restrict__ T) {
  const int lane = threadIdx.x & 31;
  const int wid  = (blockIdx.x * blockDim.x + threadIdx.x) >> 5;
  const int rt   = wid % 5;        // rank tile 0..4
  const int bs   = wid / 5;        // batch strip 0..511
  const int half = lane >> 4;      // 0: K+{0,1}, 1: K+{2,3}
  const int lm   = lane & 15;

  const float* arow = VcatT + (size_t)(rt * 16 + lm) * N_IN;  // A row (fixed M)
  const float* bcol = x + bs * 16 + lm;                       // B column (fixed N)

  v8f acc0 = {0.f, 0.f, 0.f, 0.f, 0.f, 0.f, 0.f, 0.f};
  v8f acc1 = {0.f, 0.f, 0.f, 0.f, 0.f, 0.f, 0.f, 0.f};

  #pragma unroll 2
  for (int k = 0; k < N_IN; k += 8) {
    {
      const int ka = k + 2 * half;
      v2f a = *(const v2f*)(arow + ka);
      v2f b;
      b.x = bcol[(size_t)ka * BATCH];
      b.y = bcol[(size_t)(ka + 1) * BATCH];
      acc0 = __builtin_amdgcn_wmma_f32_16x16x4_f32(false, a, false, b,
                                                   (short)0, acc0, false, false);
    }
    {
      const int ka = k + 4 + 2 * half;
      v2f a = *(const v2f*)(arow + ka);
      v2f b;
      b.x = bcol[(size_t)ka * BATCH];
      b.y = bcol[(size_t)(ka + 1) * BATCH];
      acc1 = __builtin_amdgcn_wmma_f32_16x16x4_f32(false, a, false, b,
                                                   (short)0, acc1, false, false);
    }
  }
  v8f acc = acc0 + acc1;

  // D layout: vgpr v holds row (v + 8*half), col = lane%16.
  float* trow = T + (size_t)(rt * 16 + 8 * half) * BATCH + bs * 16 + lm;
  #pragma unroll
  for (int v = 0; v < 8; ++v) trow[(size_t)v * BATCH] = acc[v];
}

// ---------------------------------------------------------------------------
// Stage 2: out = relu(Ucat @ T).  One wave = one 16x16 out tile; K=80 unrolled.
// T (2.5 MB) and Ucat (1.25 MB) stay L2/WGP$-resident; stores dominate HBM.
// ---------------------------------------------------------------------------
__global__ void __launch_bounds__(256)
elr_stage2(const float* __restrict__ Ucat, const float* __restrict__ T,
           float* __restrict__ out) {
  const int lane = threadIdx.x & 31;
  const int wid  = (blockIdx.x * blockDim.x + threadIdx.x) >> 5;
  const int nt   = wid & 511;   // batch tile 0..511
  const int mt   = wid >> 9;    // out-row tile 0..255
  const int half = lane >> 4;
  const int lm   = lane & 15;

  const float* arow = Ucat + (size_t)(mt * 16 + lm) * R_PAD;
  const float* bcol = T + nt * 16 + lm;

  v8f acc = {0.f, 0.f, 0.f, 0.f, 0.f, 0.f, 0.f, 0.f};

  #pragma unroll
  for (int k = 0; k < R_PAD; k += 4) {
    const int ka = k + 2 * half;
    v2f a = *(const v2f*)(arow + ka);   // row stride 320B, ka even -> 8B aligned
    v2f b;
    b.x = bcol[(size_t)ka * BATCH];
    b.y = bcol[(size_t)(ka + 1) * BATCH];
    acc = __builtin_amdgcn_wmma_f32_16x16x4_f32(false, a, false, b,
                                                (short)0, acc, false, false);
  }

  float* orow = out + (size_t)(mt * 16 + 8 * half) * BATCH + nt * 16 + lm;
  #pragma unroll
  for (int v = 0; v < 8; ++v) orow[(size_t)v * BATCH] = fmaxf(acc[v], 0.0f);
}

// ---------------------------------------------------------------------------
extern "C" void kernel_launch(void* const* d_in, const int* in_sizes, int n_in,
                              void* d_out, int out_size, void* d_ws, size_t ws_size,
                              hipStream_t stream) {
  const float* x     = (const float*)d_in[0];  // (N_IN, BATCH)
  const float* U     = (const float*)d_in[1];  // (N_OUT, RANK)
  const float* V     = (const float*)d_in[2];  // (N_IN, RANK)
  const float* U_res = (const float*)d_in[3];  // (N_OUT, RES_RANK)
  const float* V_res = (const float*)d_in[4];  // (N_IN, RES_RANK)
  // d_in[5] = active_cluster (selects the residual pair; the single provided
  // U_res/V_res pair is it, so the index itself is not needed).
  float* out = (float*)d_out;

  float* ws    = (float*)d_ws;
  float* VcatT = ws;                                   // R_PAD*N_IN  floats
  float* Ucat  = ws + (size_t)R_PAD * N_IN;            // N_OUT*R_PAD floats
  float* T     = Ucat + (size_t)N_OUT * R_PAD;         // R_PAD*BATCH floats
  // total ws use: (80*4096 + 4096*80 + 80*8192)*4B = 5 MB

  // prep: R_PAD*N_IN + N_OUT*R_PAD = 655360 elements -> 2560 blocks of 256
  elr_prep<<<2560, 256, 0, stream>>>(U, V, U_res, V_res, VcatT, Ucat);

  // stage1: 5 rank tiles * 512 batch strips = 2560 waves -> 320 blocks * 8 waves
  elr_stage1<<<320, 256, 0, stream>>>(x, VcatT, T);

  // stage2: 256*512 = 131072 tiles/waves -> 16384 blocks * 8 waves
  elr_stage2<<<16384, 256, 0, stream>>>(Ucat, T, out);
}